// CPA_Loss_36567351558391
// MI455X (gfx1250) — compile-verified
//
#include <hip/hip_runtime.h>
#include <hip/hip_bf16.h>

typedef __attribute__((ext_vector_type(2)))  float v2f;
typedef __attribute__((ext_vector_type(8)))  float v8f;
typedef __attribute__((ext_vector_type(4)))  unsigned int v4u;
typedef __attribute__((ext_vector_type(4)))  int v4i;
typedef __attribute__((ext_vector_type(8)))  int v8i;

#define NUM_B   16
#define NUM_C   4
#define HWPIX   65536              // 256*256
#define NPIX    (NUM_B * HWPIX)    // 1,048,576
#define SEGS    16                 // workgroups per batch image
#define CHUNK   (HWPIX / SEGS)     // 4096 pixels per WG
#define TILE    1024               // pixels per TDM subtile
#define NSUB    (CHUNK / TILE)     // 4
#define EPSF    1e-6f
#define TAUF    3.0f

// gf[t][k] compile-time constants: gf[j][k] = (c[k]/c[j])^0.8 if c[j]>c[k] else 1
// counts = [500000, 60000, 8000, 900]; column 0 is always 1.
#define GF01 0.18337664f
#define GF02 0.03658460f
#define GF03 0.00637112f
#define GF12 0.19950380f
#define GF13 0.03474340f
#define GF23 0.17414930f

// ws layout (floats)
#define WS_S1    0    // [4]  per-class sum of -log(sigma+eps)
#define WS_S2    4    //      sum of -log(sigma2+eps)
#define WS_PF    8    // [4]  proto factors
#define WS_INTER 16   // [16] dice intersection per batch
#define WS_PSUM  32   // [16] dice pred sum per batch
#define WS_TSUM  48   // [16] dice target sum per batch
#define WS_TOT   64

// ---------------------------------------------------------------------------
// Tensor Data Mover: 2D tile (tile_d0 contiguous elems x tile_d1 rows, 4B
// elements, row stride d0_stride elems) from global -> LDS (packed).
// Descriptor bitfields per CDNA5 ISA §8 (groups 0/1; groups 2/3 zero for 2D).
// ---------------------------------------------------------------------------
__device__ __forceinline__ unsigned int lds_offset(const void* p) {
    // generic LDS pointer = {shared_aperture, offset}; low 32 bits = LDS byte offset
    return (unsigned int)(unsigned long long)p;
}

__device__ __forceinline__ void tdm_load_2d(unsigned int lds_addr, const void* gptr,
                                            unsigned int tensor_d0, unsigned int tensor_d1,
                                            unsigned int tile_d0,   unsigned int tile_d1,
                                            unsigned long long d0_stride_elems) {
    unsigned long long ga = (unsigned long long)gptr;
    v4u g0;
    g0.x = 1u;                                            // count=1, user mode
    g0.y = lds_addr;                                      // lds_addr [63:32]
    g0.z = (unsigned int)(ga & 0xFFFFFFFFull);            // global_addr [95:64]
    g0.w = (unsigned int)((ga >> 32) & 0x1FFFFFFull) | (2u << 30); // [120:96] + type=2
    v8i g1;
    g1[0] = (int)(2u << 16);                              // data_size = 4B
    g1[1] = (int)((tensor_d0 & 0xFFFFu) << 16);           // tensor_dim0[15:0] @ [63:48]
    g1[2] = (int)((tensor_d0 >> 16) | ((tensor_d1 & 0xFFFFu) << 16)); // td0 hi | td1 lo
    g1[3] = (int)((tensor_d1 >> 16) | (tile_d0 << 16));   // td1 hi | tile_dim0
    g1[4] = (int)(tile_d1 & 0xFFFFu);                     // tile_dim1 | tile_dim2=0
    g1[5] = (int)(unsigned int)(d0_stride_elems & 0xFFFFFFFFull);   // dim0 stride lo
    g1[6] = (int)(unsigned int)((d0_stride_elems >> 32) & 0xFFFFull); // stride hi
    g1[7] = 0;                                            // dim1_stride = 0 (2D)
    v4i gz = {0, 0, 0, 0};
#if defined(__clang_major__) && (__clang_major__ >= 23)
    v8i gz8 = {0, 0, 0, 0, 0, 0, 0, 0};
    __builtin_amdgcn_tensor_load_to_lds(g0, g1, gz, gz, gz8, 0);
#else
    __builtin_amdgcn_tensor_load_to_lds(g0, g1, gz, gz, 0);
#endif
}

// ---------------------------------------------------------------------------
// Kernel 0: zero the workspace accumulators
// ---------------------------------------------------------------------------
__global__ void k_init(float* ws) {
    if (threadIdx.x < WS_TOT) ws[threadIdx.x] = 0.0f;
}

// ---------------------------------------------------------------------------
// Kernel 1: proto factors via WMMA f32 16x16x4 chains (diag of L*G^T, L*L^T,
// G*G^T over K=64). One wave; rows >= 4 are zero padding.
// A 16x4 layout: lanes 0-15 hold (row, K=0/1), lanes 16-31 hold (row, K=2/3);
// B 4x16 mirrors it (B[k][n] = G[n][k]). Diagonal D[j][j] -> VGPR j, lane j.
// Loads are branch-free: row clamped into bounds (row & 3 < 4 rows), value
// masked to zero for padding lanes, so EXEC stays all-ones throughout.
// ---------------------------------------------------------------------------
__global__ void k_proto(const float* __restrict__ L, const float* __restrict__ G,
                        float* __restrict__ ws) {
    const int lane = threadIdx.x;
    const int row  = lane & 15;
    const int hi   = lane >> 4;                 // 0 -> K{0,1}, 1 -> K{2,3}
    const int rowc = row & (NUM_C - 1);         // clamped, always in-bounds
    const float vm = (row < NUM_C) ? 1.0f : 0.0f;

    v8f dLG = {}; v8f dLL = {}; v8f dGG = {};
#pragma unroll
    for (int kk = 0; kk < 16; ++kk) {
        const int k0 = 4 * kk + (hi ? 2 : 0);
        v2f a, b;
        a.x = L[rowc * 64 + k0]     * vm;
        a.y = L[rowc * 64 + k0 + 1] * vm;
        b.x = G[rowc * 64 + k0]     * vm;
        b.y = G[rowc * 64 + k0 + 1] * vm;
        dLG = __builtin_amdgcn_wmma_f32_16x16x4_f32(false, a, false, b, (short)0, dLG, false, false);
        dLL = __builtin_amdgcn_wmma_f32_16x16x4_f32(false, a, false, a, (short)0, dLL, false, false);
        dGG = __builtin_amdgcn_wmma_f32_16x16x4_f32(false, b, false, b, (short)0, dGG, false, false);
    }
    if (lane < NUM_C) {
        const float lg = lane == 0 ? dLG[0] : lane == 1 ? dLG[1] : lane == 2 ? dLG[2] : dLG[3];
        const float ll = lane == 0 ? dLL[0] : lane == 1 ? dLL[1] : lane == 2 ? dLL[2] : dLL[3];
        const float gg = lane == 0 ? dGG[0] : lane == 1 ? dGG[1] : lane == 2 ? dGG[2] : dGG[3];
        const float cosv = lg / (sqrtf(ll) * sqrtf(gg) + EPSF);
        ws[WS_PF + lane] = (1.0f + TAUF) / (cosv + TAUF);
    }
}

// ---------------------------------------------------------------------------
// Kernel 2: main streaming reduction. One WG per (batch, segment). TDM
// double-buffers [4][TILE] logits + [TILE] targets tiles through LDS.
// ---------------------------------------------------------------------------
__global__ __launch_bounds__(256) void k_main(const float* __restrict__ logits,
                                              const int* __restrict__ targets,
                                              float* __restrict__ ws) {
    __shared__ float lbuf[2][NUM_C][TILE];   // 32 KiB
    __shared__ int   tbuf[2][TILE];          //  8 KiB
    __shared__ float sacc[8];

    const int tid  = threadIdx.x;
    const int b    = blockIdx.x / SEGS;
    const int seg  = blockIdx.x % SEGS;
    const int base = seg * CHUNK;                         // pixel offset in plane
    const bool issuer = (tid < 32);                       // wave 0 (uniform per wave)

    if (tid < 8) sacc[tid] = 0.0f;

    if (issuer) {
        const float* gsrc = logits + (size_t)b * NUM_C * HWPIX + base;
        const int*   tsrc = targets + (size_t)b * HWPIX + base;
        tdm_load_2d(lds_offset(&lbuf[0][0][0]), gsrc, TILE, NUM_C, TILE, NUM_C, HWPIX);
        tdm_load_2d(lds_offset(&tbuf[0][0]),    tsrc, TILE, 1,     TILE, 1,     TILE);
    }

    float s1a = 0.f, s1b = 0.f, s1c = 0.f, s1d = 0.f;
    float s2 = 0.f, iacc = 0.f, pacc = 0.f, tacc = 0.f;

    for (int s = 0; s < NSUB; ++s) {
        if (issuer) {
            if (s + 1 < NSUB) {
                const int nb = (s + 1) & 1;
                const float* gsrc = logits + (size_t)b * NUM_C * HWPIX + base + (s + 1) * TILE;
                const int*   tsrc = targets + (size_t)b * HWPIX + base + (s + 1) * TILE;
                tdm_load_2d(lds_offset(&lbuf[nb][0][0]), gsrc, TILE, NUM_C, TILE, NUM_C, HWPIX);
                tdm_load_2d(lds_offset(&tbuf[nb][0]),    tsrc, TILE, 1,     TILE, 1,     TILE);
                __builtin_amdgcn_s_wait_tensorcnt(2);  // current subtile's 2 loads done
            } else {
                __builtin_amdgcn_s_wait_tensorcnt(0);
            }
        }
        __syncthreads();
        const int cb = s & 1;
#pragma unroll
        for (int k = 0; k < TILE / 256; ++k) {
            const int p = tid + k * 256;
            const float l0 = lbuf[cb][0][p];
            const float l1 = lbuf[cb][1][p];
            const float l2 = lbuf[cb][2][p];
            const float l3 = lbuf[cb][3][p];
            const int   t  = tbuf[cb][p];

            const float m  = fmaxf(fmaxf(l0, l1), fmaxf(l2, l3));
            const float e0 = __expf(l0 - m), e1 = __expf(l1 - m);
            const float e2 = __expf(l2 - m), e3 = __expf(l3 - m);
            const float sum = e0 + e1 + e2 + e3;

            // one-hot removal: a_k = (k==t) ? 0 : e_k
            const float a0 = (t == 0) ? 0.f : e0;
            const float a1 = (t == 1) ? 0.f : e1;
            const float a2 = (t == 2) ? 0.f : e2;
            const float a3 = (t == 3) ? 0.f : e3;
            // gf row t (column 0 is always 1)
            const float g1v = (t == 0) ? GF01 : 1.0f;
            const float g2v = (t == 0) ? GF02 : (t == 1) ? GF12 : 1.0f;
            const float g3v = (t == 0) ? GF03 : (t == 1) ? GF13 : (t == 2) ? GF23 : 1.0f;
            const float et  = (t == 0) ? e0 : (t == 1) ? e1 : (t == 2) ? e2 : e3;

            const float den = fmaf(a1, g1v, fmaf(a2, g2v, fmaf(a3, g3v, a0))) + et;
            const float v1  = -__logf(et / (den + EPSF) + EPSF);
            const float v2  = -__logf(et / (sum + EPSF) + EPSF);

            s1a += (t == 0) ? v1 : 0.f;
            s1b += (t == 1) ? v1 : 0.f;
            s1c += (t == 2) ? v1 : 0.f;
            s1d += (t == 3) ? v1 : 0.f;
            s2  += v2;

            const float p1 = e1 / sum;          // softmax channel 1 (dice)
            const float tf = (float)t;
            iacc += p1 * tf;
            pacc += p1;
            tacc += tf;
        }
        __syncthreads();
    }

    // wave-level shuffle reduce, then LDS, then global atomics
    float r[8] = { s1a, s1b, s1c, s1d, s2, iacc, pacc, tacc };
#pragma unroll
    for (int q = 0; q < 8; ++q)
#pragma unroll
        for (int off = 16; off > 0; off >>= 1)
            r[q] += __shfl_down(r[q], off);
    if ((tid & 31) == 0) {
#pragma unroll
        for (int q = 0; q < 8; ++q) atomicAdd(&sacc[q], r[q]);
    }
    __syncthreads();
    if (tid == 0) {
        atomicAdd(&ws[WS_S1 + 0], sacc[0]);
        atomicAdd(&ws[WS_S1 + 1], sacc[1]);
        atomicAdd(&ws[WS_S1 + 2], sacc[2]);
        atomicAdd(&ws[WS_S1 + 3], sacc[3]);
        atomicAdd(&ws[WS_S2],     sacc[4]);
        atomicAdd(&ws[WS_INTER + b], sacc[5]);
        atomicAdd(&ws[WS_PSUM  + b], sacc[6]);
        atomicAdd(&ws[WS_TSUM  + b], sacc[7]);
    }
}

// ---------------------------------------------------------------------------
// Kernel 3: combine
// ---------------------------------------------------------------------------
__global__ void k_final(const float* __restrict__ ws, float* __restrict__ out) {
    float num = ws[WS_S2];
#pragma unroll
    for (int c = 0; c < NUM_C; ++c) num += ws[WS_PF + c] * ws[WS_S1 + c];
    const float ce = 0.5f * num / (float)NPIX;
    float dsum = 0.f;
#pragma unroll
    for (int b = 0; b < NUM_B; ++b)
        dsum += (2.0f * ws[WS_INTER + b] + 1.0f) /
                (ws[WS_PSUM + b] + ws[WS_TSUM + b] + 1.0f);
    out[0] = ce + (1.0f - dsum / (float)NUM_B);
}

extern "C" void kernel_launch(void* const* d_in, const int* in_sizes, int n_in,
                              void* d_out, int out_size, void* d_ws, size_t ws_size,
                              hipStream_t stream) {
    const float* logits  = (const float*)d_in[0];
    const int*   targets = (const int*)d_in[1];
    const float* lproto  = (const float*)d_in[2];
    const float* gproto  = (const float*)d_in[3];
    float* ws  = (float*)d_ws;
    float* out = (float*)d_out;
    (void)in_sizes; (void)n_in; (void)out_size; (void)ws_size;

    k_init <<<1, 64, 0, stream>>>(ws);
    k_proto<<<1, 32, 0, stream>>>(lproto, gproto, ws);
    k_main <<<NUM_B * SEGS, 256, 0, stream>>>(logits, targets, ws);
    k_final<<<1, 1, 0, stream>>>(ws, out);
}